// EmbPropCell_73976516706688
// MI455X (gfx1250) — compile-verified
//
#include <hip/hip_runtime.h>

typedef __attribute__((ext_vector_type(2))) float v2f;
typedef __attribute__((ext_vector_type(4))) float v4f;
typedef __attribute__((ext_vector_type(8))) float v8f;

#define DIM 128

static __device__ __forceinline__ v8f wmma_f32(v2f a, v2f b, v8f c) {
  return __builtin_amdgcn_wmma_f32_16x16x4_f32(false, a, false, b,
                                               (short)0, c, false, false);
}

// ---------------------------------------------------------------------------
// Kernel 1: out[m, :] = X[m, :] @ W^T  (pre-activation self message)
// Block = 128 threads = 4 waves. blockIdx.x -> 16-row tile.
// Wave w -> cols [32w, 32w+32): two 16-col WMMA tiles sharing the A load.
// A(16x4): row m = lane&15, k = kk + (lane>>4)*2 + {0,1}
// B(4x16): col n = lane&15, same k pattern (B[k][n] = W[n][k])
// ---------------------------------------------------------------------------
__global__ void __launch_bounds__(128)
node_selfmsg_kernel(const float* __restrict__ X, const float* __restrict__ W,
                    float* __restrict__ out, int nrows) {
  const int lane  = threadIdx.x & 31;
  const int wave  = threadIdx.x >> 5;
  const int m0    = blockIdx.x * 16;
  const int n0    = wave * 32;
  const int lo    = lane & 15;
  const int khalf = lane >> 4;

  int mrow = m0 + lo;
  if (mrow >= nrows) mrow = nrows - 1;          // clamp keeps EXEC uniform
  const float* Arow = X + (size_t)mrow * DIM;
  const float* B0   = W + (size_t)(n0 + lo) * DIM;
  const float* B1   = B0 + 16 * DIM;            // +8KB -> immediate offset

  v8f acc0 = {}, acc1 = {};
  #pragma unroll 4
  for (int kk = 0; kk < DIM; kk += 4) {
    const int k = kk + khalf * 2;
    v2f a  = *(const v2f*)(Arow + k);
    v2f b0 = *(const v2f*)(B0 + k);
    v2f b1 = *(const v2f*)(B1 + k);
    acc0 = wmma_f32(a, b0, acc0);
    acc1 = wmma_f32(a, b1, acc1);
  }
  // C/D layout: VGPR r -> row m0 + r + 8*khalf, col n0 + (lane&15)
  if (m0 + 16 <= nrows) {
    // uniform fast path: straight-line stores, no EXEC churn
    float* o = out + (size_t)(m0 + khalf * 8) * DIM + n0 + lo;
    #pragma unroll
    for (int r = 0; r < 8; ++r) {
      o[(size_t)r * DIM]      = acc0[r];
      o[(size_t)r * DIM + 16] = acc1[r];
    }
  } else {
    #pragma unroll
    for (int r = 0; r < 8; ++r) {
      const int orow = m0 + khalf * 8 + r;
      if (orow < nrows) {
        float* o = out + (size_t)orow * DIM + lo;
        o[n0]      = acc0[r];
        o[n0 + 16] = acc1[r];
      }
    }
  }
}

// ---------------------------------------------------------------------------
// Kernel 2: per 32-edge tile:
//   msg = ew * ( src_x[s] @ Wi^T  +  (src_x[s] * dst_x[d]) @ We^T )
//   atomicAdd into out[dst]. Both GEMMs fused into the same accumulators.
// Block = 128 threads = 4 waves; wave w owns col tiles {32w, 32w+16} for BOTH
// 16-edge sub-tiles: 4 accumulators, 8 WMMAs per 8 b64 loads per K-step.
// LDS row stride 132 (mod 64 = 4) -> conflict-free strided ds_load_b64.
// ---------------------------------------------------------------------------
__global__ void __launch_bounds__(128)
edge_msg_kernel(const float* __restrict__ src_x, const float* __restrict__ dst_x,
                const int* __restrict__ src_idx, const int* __restrict__ dst_idx,
                const float* __restrict__ ew,
                const float* __restrict__ Wi, const float* __restrict__ We,
                float* __restrict__ out, int nedges) {
  __shared__ float s_xs[32][132];
  __shared__ float s_p [32][132];
  __shared__ float s_ew[32];
  __shared__ int   s_dst[32];

  const int tid = threadIdx.x;
  const int e0  = blockIdx.x * 32;

  // --- stage 32 gathered rows: thread t -> edge-row t/4, 32 cols at (t%4)*32
  {
    const int row = tid >> 2;
    const int cb  = (tid & 3) * 32;
    int e = e0 + row;
    if (e >= nedges) e = nedges - 1;            // clamp (weight zeroed below)
    const float* xs = src_x + (size_t)src_idx[e] * DIM + cb;
    const float* xd = dst_x + (size_t)dst_idx[e] * DIM + cb;
    #pragma unroll
    for (int j = 0; j < 32; j += 4) {
      v4f a = *(const v4f*)(xs + j);
      v4f b = *(const v4f*)(xd + j);
      *(v4f*)(&s_xs[row][cb + j]) = a;
      *(v4f*)(&s_p [row][cb + j]) = a * b;
    }
    if (tid < 32) {
      const int e2 = e0 + tid;
      s_ew[tid]  = (e2 < nedges) ? ew[e2] : 0.0f;
      s_dst[tid] = dst_idx[(e2 < nedges) ? e2 : (nedges - 1)];
    }
  }
  __syncthreads();

  const int lane  = tid & 31;
  const int wave  = tid >> 5;
  const int n0    = wave * 32;                  // first of two 16-col tiles
  const int lo    = lane & 15;
  const int khalf = lane >> 4;

  const float* BiA = Wi + (size_t)(n0 + lo) * DIM;   // B[k][n] = W[n][k]
  const float* BeA = We + (size_t)(n0 + lo) * DIM;
  const float* BiB = BiA + 16 * DIM;                 // +8KB -> imm offset
  const float* BeB = BeA + 16 * DIM;

  const float* Ax0 = &s_xs[lo][0];       // edge tile 0 (rows 0..15)
  const float* Ap0 = &s_p [lo][0];
  const float* Ax1 = &s_xs[16 + lo][0];  // edge tile 1 (rows 16..31)
  const float* Ap1 = &s_p [16 + lo][0];

  v8f acc00 = {}, acc01 = {}, acc10 = {}, acc11 = {};
  #pragma unroll 2
  for (int kk = 0; kk < DIM; kk += 4) {
    const int k = kk + khalf * 2;
    v2f a0x = *(const v2f*)(Ax0 + k);
    v2f a0p = *(const v2f*)(Ap0 + k);
    v2f a1x = *(const v2f*)(Ax1 + k);
    v2f a1p = *(const v2f*)(Ap1 + k);
    v2f bia = *(const v2f*)(BiA + k);
    v2f bea = *(const v2f*)(BeA + k);
    v2f bib = *(const v2f*)(BiB + k);
    v2f beb = *(const v2f*)(BeB + k);
    acc00 = wmma_f32(a0x, bia, acc00);
    acc00 = wmma_f32(a0p, bea, acc00);
    acc01 = wmma_f32(a0x, bib, acc01);
    acc01 = wmma_f32(a0p, beb, acc01);
    acc10 = wmma_f32(a1x, bia, acc10);
    acc10 = wmma_f32(a1p, bea, acc10);
    acc11 = wmma_f32(a1x, bib, acc11);
    acc11 = wmma_f32(a1p, beb, acc11);
  }

  // scatter-add: VGPR r -> edge-row (tile*16 + r + 8*khalf), col n0{+16} + lo
  #pragma unroll
  for (int r = 0; r < 8; ++r) {
    const int row0 = r + 8 * khalf;
    const int row1 = 16 + row0;
    float* o0 = out + (size_t)s_dst[row0] * DIM + lo;
    float* o1 = out + (size_t)s_dst[row1] * DIM + lo;
    const float w0 = s_ew[row0];
    const float w1 = s_ew[row1];
    atomicAdd(o0 + n0,      w0 * acc00[r]);
    atomicAdd(o0 + n0 + 16, w0 * acc01[r]);
    atomicAdd(o1 + n0,      w1 * acc10[r]);
    atomicAdd(o1 + n0 + 16, w1 * acc11[r]);
  }
}

// ---------------------------------------------------------------------------
// Kernel 3: in-place leaky_relu (negative_slope = 0.01), float4 vectorized
// ---------------------------------------------------------------------------
__global__ void __launch_bounds__(256)
lrelu_kernel(float* __restrict__ out, int n4) {
  const int i = blockIdx.x * blockDim.x + threadIdx.x;
  if (i < n4) {
    v4f x = *(v4f*)(out + (size_t)i * 4);
    #pragma unroll
    for (int c = 0; c < 4; ++c) x[c] = (x[c] > 0.0f) ? x[c] : 0.01f * x[c];
    *(v4f*)(out + (size_t)i * 4) = x;
  }
}

// ---------------------------------------------------------------------------
// inputs: 0 src_x[N*128] f32, 1 dst_x[N*128] f32, 2 edge_index[2*E] i32,
//         3 edge_weight[E] f32, 4 W_intra[128*128] f32, 5 W_inter[128*128] f32
// output: [N*128] f32
// ---------------------------------------------------------------------------
extern "C" void kernel_launch(void* const* d_in, const int* in_sizes, int n_in,
                              void* d_out, int out_size, void* d_ws, size_t ws_size,
                              hipStream_t stream) {
  const float* src_x = (const float*)d_in[0];
  const float* dst_x = (const float*)d_in[1];
  const int*   eidx  = (const int*)  d_in[2];
  const float* ew    = (const float*)d_in[3];
  const float* Wi    = (const float*)d_in[4];
  const float* We    = (const float*)d_in[5];
  float*       out   = (float*)d_out;

  const int N = in_sizes[0] / DIM;
  const int E = in_sizes[2] / 2;
  const int* src_idx = eidx;        // edge_index[0]
  const int* dst_idx = eidx + E;    // edge_index[1]

  // 1) out = dst_x @ W_intra^T   (initializes accumulator)
  node_selfmsg_kernel<<<(N + 15) / 16, 128, 0, stream>>>(dst_x, Wi, out, N);
  // 2) fused per-edge messages, atomically scattered into out
  edge_msg_kernel<<<(E + 31) / 32, 128, 0, stream>>>(src_x, dst_x, src_idx, dst_idx,
                                                     ew, Wi, We, out, E);
  // 3) leaky_relu in place
  const int n4 = (N * DIM) / 4;
  lrelu_kernel<<<(n4 + 255) / 256, 256, 0, stream>>>(out, n4);
}